// GATExpert_2310692405503
// MI455X (gfx1250) — compile-verified
//
#include <hip/hip_runtime.h>
#include <hip/hip_bf16.h>
#include <math.h>

typedef __attribute__((ext_vector_type(2))) float v2f;
typedef __attribute__((ext_vector_type(8))) float v8f;

#define NEG_SLOPE 0.2f
#define BN_EPS 1e-5f

// ---- order-preserving float <-> uint encoding for atomic segment-max ----
__device__ __forceinline__ unsigned encodeF(float f) {
    unsigned u = __float_as_uint(f);
    return (u & 0x80000000u) ? ~u : (u | 0x80000000u);
}
__device__ __forceinline__ float decodeF(unsigned u) {
    unsigned b = (u & 0x80000000u) ? (u & 0x7fffffffu) : ~u;
    return __uint_as_float(b);
}
__device__ __forceinline__ float lrelu(float v) {
    return v > 0.0f ? v : NEG_SLOPE * v;
}

// ---- generic fill ----
__global__ __launch_bounds__(256) void fill_u32(unsigned* __restrict__ p, unsigned v, int n) {
    int i = blockIdx.x * blockDim.x + threadIdx.x;
    if (i < n) p[i] = v;
}

// ---- out[n*C + c] = bias[c] ----
__global__ __launch_bounds__(256) void init_out_bias(float* __restrict__ out,
                                                     const float* __restrict__ b,
                                                     int total, int C) {
    int i = blockIdx.x * blockDim.x + threadIdx.x;
    if (i < total) out[i] = b[i % C];
}

// ---- fp32 WMMA GEMM: C[M,Ncols] = A[M,K] @ B[K,Ncols]; M,Ncols multiples of 16, K multiple of 4
// One wave per 16x16 output tile. V_WMMA_F32_16X16X4_F32 lane layouts per CDNA5 ISA 7.12.2:
//   A: M = lane&15, K = v + 2*(lane>>4)   (2 VGPRs)
//   B: N = lane&15, K = v + 2*(lane>>4)   (2 VGPRs)
//   C: N = lane&15, M = v + 8*(lane>>4)   (8 VGPRs)
__global__ __launch_bounds__(256) void wmma_gemm_f32(const float* __restrict__ A,
                                                     const float* __restrict__ B,
                                                     float* __restrict__ C,
                                                     int M, int K, int Ncols) {
    int wave = (int)((blockIdx.x * (long long)blockDim.x + threadIdx.x) >> 5);
    int lane = threadIdx.x & 31;
    int ntN = Ncols >> 4;
    int totalWaves = (M >> 4) * ntN;
    if (wave >= totalWaves) return;
    int mt = wave / ntN, nt = wave - mt * ntN;
    int m0 = mt << 4, n0 = nt << 4;
    int lo = lane & 15;
    int hi = lane >> 4;
    const float* arow = A + (size_t)(m0 + lo) * K;   // this lane's A row
    int ncol = n0 + lo;                              // this lane's B/C column

    v8f acc = {0.f, 0.f, 0.f, 0.f, 0.f, 0.f, 0.f, 0.f};
    for (int k0 = 0; k0 < K; k0 += 4) {
        int kb = k0 + 2 * hi;
        v2f a;
        a.x = arow[kb];
        a.y = arow[kb + 1];
        v2f b;
        b.x = B[(size_t)kb * Ncols + ncol];
        b.y = B[(size_t)(kb + 1) * Ncols + ncol];
        acc = __builtin_amdgcn_wmma_f32_16x16x4_f32(
            /*neg_a=*/false, a, /*neg_b=*/false, b,
            /*c_mod=*/(short)0, acc, /*reuse_a=*/false, /*reuse_b=*/false);
    }
#pragma unroll
    for (int v = 0; v < 8; ++v) {
        int row = m0 + v + 8 * hi;
        C[(size_t)row * Ncols + ncol] = acc[v];
    }
}

// ---- per-node attention coefficients: as[n,h] = <h[n,h,:], a_src[h,:]> ----
__global__ __launch_bounds__(256) void alpha_kernel(const float* __restrict__ hf,
                                                    const float* __restrict__ a_s,
                                                    const float* __restrict__ a_d,
                                                    float* __restrict__ as_o,
                                                    float* __restrict__ ad_o,
                                                    int N, int H, int Cc) {
    int n = blockIdx.x * blockDim.x + threadIdx.x;
    if (n >= N) return;
    int F = H * Cc;
    for (int h = 0; h < H; ++h) {
        float s = 0.f, d = 0.f;
        for (int c = 0; c < Cc; ++c) {
            float v = hf[(size_t)n * F + h * Cc + c];
            s += v * a_s[h * Cc + c];
            d += v * a_d[h * Cc + c];
        }
        as_o[n * H + h] = s;
        ad_o[n * H + h] = d;
    }
}

// ---- edge pass 1: segment max (atomicMax on encoded uint) ----
__global__ __launch_bounds__(256) void edge_max(const int* __restrict__ ei, int E, int N,
                                                const float* __restrict__ as_i,
                                                const float* __restrict__ ad_i,
                                                unsigned* __restrict__ menc, int H) {
    int e = blockIdx.x * blockDim.x + threadIdx.x;
    int Etot = E + N;
    if (e >= Etot) return;
    int s = (e < E) ? ei[e] : (e - E);
    int d = (e < E) ? ei[E + e] : (e - E);
    for (int h = 0; h < H; ++h) {
        float v = lrelu(as_i[s * H + h] + ad_i[d * H + h]);
        atomicMax(&menc[d * H + h], encodeF(v));
    }
}

// ---- edge pass 2: segment sum of exp(e - m) ----
__global__ __launch_bounds__(256) void edge_sum(const int* __restrict__ ei, int E, int N,
                                                const float* __restrict__ as_i,
                                                const float* __restrict__ ad_i,
                                                const unsigned* __restrict__ menc,
                                                float* __restrict__ z, int H) {
    int e = blockIdx.x * blockDim.x + threadIdx.x;
    int Etot = E + N;
    if (e >= Etot) return;
    int s = (e < E) ? ei[e] : (e - E);
    int d = (e < E) ? ei[E + e] : (e - E);
    for (int h = 0; h < H; ++h) {
        float v = lrelu(as_i[s * H + h] + ad_i[d * H + h]);
        float ee = __expf(v - decodeF(menc[d * H + h]));
        atomicAdd(&z[d * H + h], ee);
    }
}

// ---- edge pass 3: out[dst] += alpha * h[src]; one thread per (edge, channel) ----
__global__ __launch_bounds__(256) void edge_agg(const int* __restrict__ ei, int E, int N,
                                                const float* __restrict__ as_i,
                                                const float* __restrict__ ad_i,
                                                const unsigned* __restrict__ menc,
                                                const float* __restrict__ z,
                                                const float* __restrict__ hsrc,
                                                float* __restrict__ out, int H, int Cc) {
    long long t = blockIdx.x * (long long)blockDim.x + threadIdx.x;
    long long Etot = (long long)E + N;
    if (t >= Etot * Cc) return;
    int e = (int)(t / Cc);
    int c = (int)(t - (long long)e * Cc);
    int s = (e < E) ? ei[e] : (e - E);
    int d = (e < E) ? ei[E + e] : (e - E);
    int F = H * Cc;
    for (int h = 0; h < H; ++h) {
        float v = lrelu(as_i[s * H + h] + ad_i[d * H + h]);
        float al = __expf(v - decodeF(menc[d * H + h])) / (z[d * H + h] + 1e-16f);
        float msg = hsrc[(size_t)s * F + h * Cc + c] * al;
        atomicAdd(&out[(size_t)d * F + h * Cc + c], msg);
    }
}

// ---- bias + batchnorm(eval) + ELU, writes into a (possibly aliased-free) buffer ----
__global__ __launch_bounds__(256) void bn_elu(const float* __restrict__ agg,
                                              const float* __restrict__ b1,
                                              const float* __restrict__ g,
                                              const float* __restrict__ be,
                                              const float* __restrict__ mu,
                                              const float* __restrict__ var,
                                              float* __restrict__ outp, int total, int F) {
    int i = blockIdx.x * blockDim.x + threadIdx.x;
    if (i >= total) return;
    int j = i % F;
    float v = agg[i] + b1[j];
    v = (v - mu[j]) * g[j] * rsqrtf(var[j] + BN_EPS) + be[j];
    outp[i] = v > 0.f ? v : (__expf(v) - 1.f);
}

static inline int nblk(long long t) { return (int)((t + 255) / 256); }

extern "C" void kernel_launch(void* const* d_in, const int* in_sizes, int n_in,
                              void* d_out, int out_size, void* d_ws, size_t ws_size,
                              hipStream_t stream) {
    const float* x      = (const float*)d_in[0];
    const int*   ei     = (const int*)d_in[1];
    const float* W1     = (const float*)d_in[2];
    const float* a_src1 = (const float*)d_in[3];
    const float* a_dst1 = (const float*)d_in[4];
    const float* b1     = (const float*)d_in[5];
    const float* bn_g   = (const float*)d_in[6];
    const float* bn_b   = (const float*)d_in[7];
    const float* bn_m   = (const float*)d_in[8];
    const float* bn_v   = (const float*)d_in[9];
    const float* W2     = (const float*)d_in[10];
    const float* a_src2 = (const float*)d_in[11];
    const float* a_dst2 = (const float*)d_in[12];
    const float* b2     = (const float*)d_in[13];

    const int IN_DIM = 128, F1 = 256, C1 = 64, H1 = 4, C2 = 64;
    int N = in_sizes[0] / IN_DIM;   // 50000
    int E = in_sizes[1] / 2;        // 800000
    int Etot = E + N;               // self-loops appended

    float* ws = (float*)d_ws;
    size_t o = 0;
    float*    h1   = ws + o; o += (size_t)N * F1;   // layer-1 features; reused for BN/ELU output
    float*    agg1 = ws + o; o += (size_t)N * F1;   // layer-1 aggregation
    float*    as1  = ws + o; o += (size_t)N * H1;
    float*    ad1  = ws + o; o += (size_t)N * H1;
    unsigned* m1   = (unsigned*)(ws + o); o += (size_t)N * H1;
    float*    z1   = ws + o; o += (size_t)N * H1;
    float*    h2   = ws + o; o += (size_t)N * C2;
    float*    as2  = ws + o; o += (size_t)N;
    float*    ad2  = ws + o; o += (size_t)N;
    unsigned* m2   = (unsigned*)(ws + o); o += (size_t)N;
    float*    z2   = ws + o; o += (size_t)N;

    // zero-init accumulators (0u bits == 0.0f; 0u is below every encoded float)
    fill_u32<<<nblk((long long)N * F1), 256, 0, stream>>>((unsigned*)agg1, 0u, N * F1);
    fill_u32<<<nblk(N * H1), 256, 0, stream>>>(m1, 0u, N * H1);
    fill_u32<<<nblk(N * H1), 256, 0, stream>>>((unsigned*)z1, 0u, N * H1);
    fill_u32<<<nblk(N), 256, 0, stream>>>(m2, 0u, N);
    fill_u32<<<nblk(N), 256, 0, stream>>>((unsigned*)z2, 0u, N);

    // ---- layer 1 ----
    {
        long long waves = (long long)(N / 16) * (F1 / 16);
        wmma_gemm_f32<<<nblk(waves * 32), 256, 0, stream>>>(x, W1, h1, N, IN_DIM, F1);
    }
    alpha_kernel<<<nblk(N), 256, 0, stream>>>(h1, a_src1, a_dst1, as1, ad1, N, H1, C1);
    edge_max<<<nblk(Etot), 256, 0, stream>>>(ei, E, N, as1, ad1, m1, H1);
    edge_sum<<<nblk(Etot), 256, 0, stream>>>(ei, E, N, as1, ad1, m1, z1, H1);
    edge_agg<<<nblk((long long)Etot * C1), 256, 0, stream>>>(ei, E, N, as1, ad1, m1, z1, h1,
                                                             agg1, H1, C1);
    // bias + BN + ELU, reusing h1 as the layer-2 input buffer
    bn_elu<<<nblk((long long)N * F1), 256, 0, stream>>>(agg1, b1, bn_g, bn_b, bn_m, bn_v, h1,
                                                        N * F1, F1);

    // ---- layer 2 ----
    {
        long long waves = (long long)(N / 16) * (C2 / 16);
        wmma_gemm_f32<<<nblk(waves * 32), 256, 0, stream>>>(h1, W2, h2, N, F1, C2);
    }
    alpha_kernel<<<nblk(N), 256, 0, stream>>>(h2, a_src2, a_dst2, as2, ad2, N, 1, C2);
    edge_max<<<nblk(Etot), 256, 0, stream>>>(ei, E, N, as2, ad2, m2, 1);
    edge_sum<<<nblk(Etot), 256, 0, stream>>>(ei, E, N, as2, ad2, m2, z2, 1);
    init_out_bias<<<nblk((long long)N * C2), 256, 0, stream>>>((float*)d_out, b2, N * C2, C2);
    edge_agg<<<nblk((long long)Etot * C2), 256, 0, stream>>>(ei, E, N, as2, ad2, m2, z2, h2,
                                                             (float*)d_out, 1, C2);
}